// JSONTreeLSTM_23132693856446
// MI455X (gfx1250) — compile-verified
//
#include <hip/hip_runtime.h>
#include <hip/hip_bf16.h>

typedef float v8f   __attribute__((ext_vector_type(8)));
typedef float f32x2 __attribute__((ext_vector_type(2)));

#define KROWS 8192
#define LSEQ  128
#define DH    64
#define NG    256   // 4*DH gate width

__device__ __forceinline__ float sigf(float x)       { return 1.0f / (1.0f + __expf(-x)); }
__device__ __forceinline__ float tanhf_fast(float x) { return 1.0f - 2.0f / (__expf(2.0f * x) + 1.0f); }

// 8 output tiles (one 16x128 slab of gates) : acc[n] += A(16x4,kc) * B(4x16,kc,n)
__device__ __forceinline__ void gemm8(const float* Abase, const float* Bbase, v8f acc[8]) {
    #pragma unroll
    for (int kc = 0; kc < 16; ++kc) {
        f32x2 A = *(const f32x2*)(Abase + kc * 4);
        f32x2 B[8];
        #pragma unroll
        for (int n = 0; n < 8; ++n)
            B[n] = *(const f32x2*)(Bbase + n * 1024 + kc * 4);
        #pragma unroll
        for (int n = 0; n < 8; ++n)
            acc[n] = __builtin_amdgcn_wmma_f32_16x16x4_f32(
                false, A, false, B[n], (short)0, acc[n], false, false);
    }
}

__device__ __forceinline__ void gemm4(const float* Abase, const float* Bbase, v8f acc[4]) {
    #pragma unroll
    for (int kc = 0; kc < 16; ++kc) {
        f32x2 A = *(const f32x2*)(Abase + kc * 4);
        f32x2 B[4];
        #pragma unroll
        for (int n = 0; n < 4; ++n)
            B[n] = *(const f32x2*)(Bbase + n * 1024 + kc * 4);
        #pragma unroll
        for (int n = 0; n < 4; ++n)
            acc[n] = __builtin_amdgcn_wmma_f32_16x16x4_f32(
                false, A, false, B[n], (short)0, acc[n], false, false);
    }
}

// ---------------------------------------------------------------------------
// Kernel 0: fold the rank-1 input path into u,v vectors; zero accumulators.
// ws layout (floats): [0:256]=u  [256:512]=v  [512:576]=fc_acc  [576:640]=hs_acc
// ---------------------------------------------------------------------------
__global__ void __launch_bounds__(256, 1)
jtl_setup_kernel(const float* __restrict__ W_ih,
                 const float* __restrict__ W_num,
                 const float* __restrict__ b_num,
                 const float* __restrict__ b_ih,
                 const float* __restrict__ b_hh,
                 float* __restrict__ ws) {
    int g = threadIdx.x;                 // 256 threads
    float u = 0.0f, v = 0.0f;
    #pragma unroll 8
    for (int k = 0; k < DH; ++k) {
        float w = W_ih[g * (2 * DH) + k];
        u = fmaf(w, W_num[k], u);
        v = fmaf(w, b_num[k], v);
    }
    ws[g]       = u;
    ws[256 + g] = v + b_ih[g] + b_hh[g];
    if (g < 2 * DH) ws[512 + g] = 0.0f;
}

// ---------------------------------------------------------------------------
// Kernel 1: LSTM scan + reduction head.
// 64 blocks x 256 threads (8 waves); each wave owns 16 rows of K.
// Dynamic LDS floats: [0:16384)=W_hh  [16384:24576)=per-wave h tiles
//                     [24576:25088)=interleaved (u,v)
// __launch_bounds__(256, 1): occupancy is grid-limited anyway (512 waves
// device-wide); give the allocator the full register file so the unrolled
// WMMA pipeline never touches scratch.
// ---------------------------------------------------------------------------
__global__ void __launch_bounds__(256, 1)
jtl_scan_kernel(const float* __restrict__ x,
                const float* __restrict__ W_hh,
                const float* __restrict__ W_aout,
                const float* __restrict__ b_aout,
                const float* __restrict__ W_fh,
                const float* __restrict__ b_fh,
                const float* __restrict__ ws) {
    extern __shared__ float smem[];
    float* wlds  = smem;                 // 16384 floats
    float* hlds  = smem + 16384;         // 8192 floats
    float* uvlds = smem + 24576;         // 512 floats

    const int tid  = threadIdx.x;
    const int wave = tid >> 5;
    const int lane = tid & 31;
    const int half = lane >> 4;
    const int ln16 = lane & 15;
    float* hbuf = hlds + wave * 1024;
    const int rowbase = blockIdx.x * 128 + wave * 16;

    float* fc_acc = (float*)(ws + 512);
    float* hs_acc = (float*)(ws + 576);

    for (int i = tid; i < NG * DH; i += 256) wlds[i] = W_hh[i];
    for (int i = tid; i < 512; i += 256)
        uvlds[i] = (i & 1) ? ws[256 + (i >> 1)] : ws[i >> 1];
    for (int i = lane; i < 1024; i += 32) hbuf[i] = 0.0f;

    // Loop-invariant per-lane base pointers; all inner offsets are constants.
    const float* Abase = hbuf + ln16 * DH + half * 2;        // + kc*4
    const float* Bbase = wlds + ln16 * DH + half * 2;        // + n*1024 + kc*4
    const float* UVb   = uvlds + ln16 * 2;                   // + n*32
    float*       Hst   = hbuf + half * 8 * DH + ln16;        // + j*64 + q*16
    const float* xbase = x + (size_t)(rowbase + 8 * half) * LSEQ;  // + j*128 + t

    v8f cst[4];
    #pragma unroll
    for (int q = 0; q < 4; ++q)
        #pragma unroll
        for (int j = 0; j < 8; ++j) cst[q][j] = 0.0f;

    __syncthreads();

    for (int t = 0; t < LSEQ; ++t) {
        float xv[8];
        #pragma unroll
        for (int j = 0; j < 8; ++j) xv[j] = xbase[j * LSEQ + t];

        v8f acc[8];
        v8f si[4];

        // ---- pass 1: i (tiles 0-3), f (tiles 4-7) ----
        #pragma unroll
        for (int n = 0; n < 8; ++n) {
            f32x2 uv = *(const f32x2*)(UVb + n * 32);
            #pragma unroll
            for (int j = 0; j < 8; ++j) acc[n][j] = fmaf(xv[j], uv[0], uv[1]);
        }
        gemm8(Abase, Bbase, acc);
        #pragma unroll
        for (int q = 0; q < 4; ++q)
            #pragma unroll
            for (int j = 0; j < 8; ++j) {
                si[q][j]  = sigf(acc[q][j]);
                cst[q][j] = sigf(acc[4 + q][j]) * cst[q][j];
            }

        // ---- pass 2: g (tiles 8-11), o (tiles 12-15) ----
        #pragma unroll
        for (int n = 0; n < 8; ++n) {
            f32x2 uv = *(const f32x2*)(UVb + (8 + n) * 32);
            #pragma unroll
            for (int j = 0; j < 8; ++j) acc[n][j] = fmaf(xv[j], uv[0], uv[1]);
        }
        gemm8(Abase, Bbase + 8 * 1024, acc);
        #pragma unroll
        for (int q = 0; q < 4; ++q)
            #pragma unroll
            for (int j = 0; j < 8; ++j) {
                float c2 = fmaf(si[q][j], tanhf_fast(acc[q][j]), cst[q][j]);
                float h2 = sigf(acc[4 + q][j]) * tanhf_fast(c2);
                cst[q][j] = c2;
                Hst[j * DH + q * 16] = h2;
            }
        __syncthreads();   // orders LDS h store -> next-step A load
    }

    // ---- head: h_hat = h @ W_aout^T + b_aout ; f = h_hat @ W_fh^T + b_fh ----
    for (int i = tid; i < DH * DH; i += 256) {
        wlds[i]           = W_aout[i];
        wlds[DH * DH + i] = W_fh[i];
    }
    __syncthreads();

    v8f hh[4];
    #pragma unroll
    for (int n = 0; n < 4; ++n) {
        float b = b_aout[n * 16 + ln16];
        #pragma unroll
        for (int j = 0; j < 8; ++j) hh[n][j] = b;
    }
    gemm4(Abase, Bbase, hh);

    #pragma unroll
    for (int n = 0; n < 4; ++n) {
        float s = 0.0f;
        #pragma unroll
        for (int j = 0; j < 8; ++j) {
            s += hh[n][j];
            Hst[j * DH + n * 16] = hh[n][j];
        }
        atomicAdd(hs_acc + n * 16 + ln16, s);
    }
    __syncthreads();

    v8f ff[4];
    #pragma unroll
    for (int n = 0; n < 4; ++n) {
        float b = b_fh[n * 16 + ln16];
        #pragma unroll
        for (int j = 0; j < 8; ++j) ff[n][j] = b;
    }
    gemm4(Abase, Bbase + DH * DH, ff);

    #pragma unroll
    for (int n = 0; n < 4; ++n) {
        float s = 0.0f;
        #pragma unroll
        for (int j = 0; j < 8; ++j) s = fmaf(sigf(ff[n][j]), cst[n][j], s);
        atomicAdd(fc_acc + n * 16 + ln16, s);
    }
}

// ---------------------------------------------------------------------------
// Kernel 2: the (1 x D) tail. 64 threads.
// ---------------------------------------------------------------------------
__global__ void __launch_bounds__(64, 1)
jtl_final_kernel(const float* __restrict__ W_iouh,
                 const float* __restrict__ b_iouh,
                 const float* __restrict__ W_oout,
                 const float* __restrict__ b_oout,
                 const float* __restrict__ ws,
                 float* __restrict__ out) {
    __shared__ float hs[DH], hobj[DH];
    int j = threadIdx.x;
    hs[j] = ws[576 + j];
    __syncthreads();
    float di = b_iouh[j], dO = b_iouh[DH + j], du = b_iouh[2 * DH + j];
    #pragma unroll 8
    for (int k = 0; k < DH; ++k) {
        float h = hs[k];
        di = fmaf(h, W_iouh[j * DH + k], di);
        dO = fmaf(h, W_iouh[(DH + j) * DH + k], dO);
        du = fmaf(h, W_iouh[(2 * DH + j) * DH + k], du);
    }
    float c_obj = sigf(di) * tanhf_fast(du) + ws[512 + j];
    float h_obj = sigf(dO) * tanhf_fast(c_obj);
    out[DH + j] = c_obj;
    hobj[j] = h_obj;
    __syncthreads();
    float a = b_oout[j];
    #pragma unroll 8
    for (int k = 0; k < DH; ++k) a = fmaf(hobj[k], W_oout[j * DH + k], a);
    out[j] = a;
}

extern "C" void kernel_launch(void* const* d_in, const int* in_sizes, int n_in,
                              void* d_out, int out_size, void* d_ws, size_t ws_size,
                              hipStream_t stream) {
    const float* x      = (const float*)d_in[0];
    const float* W_num  = (const float*)d_in[1];
    const float* b_num  = (const float*)d_in[2];
    const float* W_ih   = (const float*)d_in[3];
    const float* W_hh   = (const float*)d_in[4];
    const float* b_ih   = (const float*)d_in[5];
    const float* b_hh   = (const float*)d_in[6];
    const float* W_aout = (const float*)d_in[7];
    const float* b_aout = (const float*)d_in[8];
    const float* W_fh   = (const float*)d_in[9];
    const float* b_fh   = (const float*)d_in[10];
    const float* W_iouh = (const float*)d_in[11];
    const float* b_iouh = (const float*)d_in[12];
    const float* W_oout = (const float*)d_in[13];
    const float* b_oout = (const float*)d_in[14];
    float* ws  = (float*)d_ws;
    float* out = (float*)d_out;

    jtl_setup_kernel<<<1, 256, 0, stream>>>(W_ih, W_num, b_num, b_ih, b_hh, ws);
    jtl_scan_kernel<<<KROWS / 128, 256, 25088 * sizeof(float), stream>>>(
        x, W_hh, W_aout, b_aout, W_fh, b_fh, ws);
    jtl_final_kernel<<<1, DH, 0, stream>>>(W_iouh, b_iouh, W_oout, b_oout, ws, out);
}